// Slater_42125039239217
// MI455X (gfx1250) — compile-verified
//
#include <hip/hip_runtime.h>
#include <hip/hip_bf16.h>

#ifndef __has_builtin
#define __has_builtin(x) 0
#endif

#if __has_builtin(__builtin_amdgcn_global_load_async_to_lds_b128)
#define HAVE_ASYNC_LDS 1
#else
#define HAVE_ASYNC_LDS 0
#endif

#if __has_builtin(__builtin_amdgcn_wmma_f32_16x16x4_f32)
#define HAVE_WMMA_F32 1
#else
#define HAVE_WMMA_F32 0
#endif

#define N_SITES 128
#define N_EL    32
#define N_DET   16
// 36-float row stride: 16B alignment for b128, gcd(36,64)=4 softens bank conflicts.
#define LDS_STRIDE 36
#define SCR_STRIDE 36
#define SAMPLES_PER_BLOCK 32

typedef float f32x4 __attribute__((ext_vector_type(4)));
typedef float v2f   __attribute__((ext_vector_type(2)));
typedef float v8f   __attribute__((ext_vector_type(8)));
typedef int   v4i   __attribute__((vector_size(16)));

typedef __attribute__((address_space(1))) v4i* glob_v4i_ptr;
typedef __attribute__((address_space(3))) v4i* lds_v4i_ptr;

__device__ __forceinline__ float lane_bcast(float v, int srclane) {
  return __builtin_bit_cast(float,
      __builtin_amdgcn_readlane(__builtin_bit_cast(int, v), srclane));
}

// 1-ulp hardware reciprocal (v_rcp_f32): the right multiplier path for LU
// feeding a log-determinant; avoids the ~12-op IEEE div expansion per step.
__device__ __forceinline__ float fast_rcp(float x) {
#if __has_builtin(__builtin_amdgcn_rcpf)
  return __builtin_amdgcn_rcpf(x);
#else
  return 1.0f / x;
#endif
}

// One wave factors one 32x32 matrix: lane i owns row i in VGPRs.
// Blocked (16) right-looking LU; the rank-16 Schur update of the trailing
// 16x16 block runs on the matrix pipe via V_WMMA_F32_16X16X4_F32 (XDL ops
// co-execute with VALU on CDNA5), the triangular recurrences stay on VALU.
__global__ void __launch_bounds__(256)
slater_logdet_kernel(const float* __restrict__ U_up,
                     const float* __restrict__ U_dn,
                     const int*   __restrict__ y,
                     float* __restrict__ ws_log,
                     float* __restrict__ ws_sign,
                     int batch)
{
  // [0 .. 9216)      : staged U_up[d], U_dn[d]   (2*128 rows, stride 36)
  // [9216 .. 18432)  : per-wave 32x32 WMMA scratch tiles (stride 36)
  __shared__ __align__(16) float ldsAll[2 * N_SITES * LDS_STRIDE + 8 * 32 * SCR_STRIDE];

  const int tid  = threadIdx.x;
  const int lane = tid & 31;
  const int wave = tid >> 5;
  const int d    = (int)(blockIdx.x % N_DET);
  const int base = (int)(blockIdx.x / N_DET) * SAMPLES_PER_BLOCK;

  float* ldsU = ldsAll;
  float* scr  = ldsAll + 2 * N_SITES * LDS_STRIDE + wave * 32 * SCR_STRIDE;

  // ---- Stage U_up[d] and U_dn[d] into LDS via CDNA5 async-to-LDS DMA. ----
  {
    const float* upsrc = U_up + (size_t)d * (N_SITES * N_EL);
    const float* dnsrc = U_dn + (size_t)d * (N_SITES * N_EL);
    for (int c = tid; c < 2 * N_SITES * (N_EL / 4); c += 256) {
      const int spin = c >> 10;
      const int rc   = c & 1023;
      const int s    = rc >> 3;
      const int q    = rc & 7;
      const float* src = (spin ? dnsrc : upsrc) + s * N_EL + q * 4;
      float* dst = &ldsU[spin * N_SITES * LDS_STRIDE + s * LDS_STRIDE + q * 4];
#if HAVE_ASYNC_LDS
      __builtin_amdgcn_global_load_async_to_lds_b128(
          (glob_v4i_ptr)(src), (lds_v4i_ptr)(dst), 0, 0);
#else
      *(f32x4*)dst = *(const f32x4*)src;
#endif
    }
#if HAVE_ASYNC_LDS
# if __has_builtin(__builtin_amdgcn_s_wait_asynccnt)
    __builtin_amdgcn_s_wait_asynccnt(0);
# else
    asm volatile("s_wait_asynccnt 0" ::: "memory");
# endif
#endif
  }
  __syncthreads();

  const int lh  = lane >> 4;   // half-wave id
  const int l15 = lane & 15;

  for (int t = 0; t < SAMPLES_PER_BLOCK / 8; ++t) {
    const int sample = base + wave + t * 8;
    if (sample >= batch) break;   // wave-uniform

    float    ldsum = 0.0f;       // accumulated in log2 domain
    unsigned sgn   = 0u;

    for (int spin = 0; spin < 2; ++spin) {
      const int site = y[(size_t)sample * (2 * N_EL) + spin * N_EL + lane];
      const float* row = &ldsU[spin * N_SITES * LDS_STRIDE + site * LDS_STRIDE];
      float a[N_EL];
#pragma unroll
      for (int q = 0; q < N_EL / 4; ++q) {
        f32x4 v = *(const f32x4*)(row + 4 * q);
        a[4*q+0] = v[0]; a[4*q+1] = v[1]; a[4*q+2] = v[2]; a[4*q+3] = v[3];
      }

      // ---- Phase A: panel LU, k = 0..15, columns k+1..15, all rows. ----
      // a[k] is overwritten with the NEGATED multiplier (-A[i][k]/piv); this
      // makes every later update a bare v_fma and pre-negates L21 so the
      // Schur WMMA needs no operand negation (F32 WMMA has no A/B NEG).
#pragma unroll
      for (int k = 0; k < 16; ++k) {
        const float piv = lane_bcast(a[k], k);
        ldsum += __log2f(__builtin_fabsf(piv));
        sgn   ^= (__builtin_bit_cast(unsigned, piv) >> 31);
        const float npinv = -fast_rcp(piv);                          // uniform
        const float lmneg = (lane > k) ? a[k] * npinv : 0.0f;
        a[k] = lmneg;
#pragma unroll
        for (int j = k + 1; j < 16; ++j) {
          const float u = lane_bcast(a[j], k);
          a[j] = __builtin_fmaf(lmneg, u, a[j]);
        }
      }

      // ---- Phase B: U12 = L11^{-1} * A12 (rows < 16 only when the WMMA
      // Schur path exists; otherwise all rows update -> unblocked LU). ----
#pragma unroll
      for (int k = 0; k < 16; ++k) {
#if HAVE_WMMA_F32
        const float lm = (lane < 16) ? a[k] : 0.0f;  // defer rows>=16 to WMMA
#else
        const float lm = a[k];                        // a[k]==0 for lane<=k
#endif
#pragma unroll
        for (int j = 16; j < 32; ++j) {
          const float u = lane_bcast(a[j], k);
          a[j] = __builtin_fmaf(lm, u, a[j]);
        }
      }

#if HAVE_WMMA_F32
      // ---- Phase C: A22' = A22 + (-L21)*U12 on the matrix pipe. ----
      {
        // Dump the wave's 32x32 tile (rows = lanes) to per-wave LDS scratch.
        float* rb = scr + lane * SCR_STRIDE;
#pragma unroll
        for (int q = 0; q < 8; ++q) {
          f32x4 v; v[0]=a[4*q+0]; v[1]=a[4*q+1]; v[2]=a[4*q+2]; v[3]=a[4*q+3];
          *(f32x4*)(rb + 4 * q) = v;
        }
        // C operand: 32-bit 16x16 C layout -> VGPR v holds rows v / v+8 by half.
        const float* cb = scr + (16 + 8 * lh) * SCR_STRIDE + 16 + l15;
        v8f c;
#pragma unroll
        for (int v = 0; v < 8; ++v) c[v] = cb[v * SCR_STRIDE];

#pragma unroll
        for (int kk = 0; kk < 16; kk += 4) {
          // A operand (16x4, -L21 chunk): lane m: K=kk,kk+1; lane m+16: K=kk+2,kk+3.
          const float* ab = scr + (16 + l15) * SCR_STRIDE + 2 * lh + kk;
          v2f av; av[0] = ab[0]; av[1] = ab[1];          // one ds_load_b64
          // B operand (4x16, U12 chunk): VGPR0 row kk(+2), VGPR1 row kk+1(+3).
          const float* bb = scr + (2 * lh + kk) * SCR_STRIDE + 16 + l15;
          v2f bv; bv[0] = bb[0]; bv[1] = bb[SCR_STRIDE];
          c = __builtin_amdgcn_wmma_f32_16x16x4_f32(
                  false, av, false, bv, (short)0, c, false, false);
        }

        // Write D back, then reload trailing rows into row-per-lane layout.
        float* db = scr + (16 + 8 * lh) * SCR_STRIDE + 16 + l15;
#pragma unroll
        for (int v = 0; v < 8; ++v) db[v * SCR_STRIDE] = c[v];

        const float* rr = scr + lane * SCR_STRIDE + 16;  // lanes>=16: A22' rows
#pragma unroll
        for (int q = 0; q < 4; ++q) {
          f32x4 v = *(const f32x4*)(rr + 4 * q);
          a[16+4*q+0] = v[0]; a[16+4*q+1] = v[1];
          a[16+4*q+2] = v[2]; a[16+4*q+3] = v[3];
        }
      }
#endif

      // ---- Phase D: LU of the trailing 16x16 block (lanes 16..31). ----
#pragma unroll
      for (int k = 16; k < 32; ++k) {
        const float piv = lane_bcast(a[k], k);
        ldsum += __log2f(__builtin_fabsf(piv));
        sgn   ^= (__builtin_bit_cast(unsigned, piv) >> 31);
        const float npinv = -fast_rcp(piv);
        const float lmneg = (lane > k) ? a[k] * npinv : 0.0f;
#pragma unroll
        for (int j = k + 1; j < 32; ++j) {
          const float u = lane_bcast(a[j], k);
          a[j] = __builtin_fmaf(lmneg, u, a[j]);
        }
      }
    }

    if (lane == 0) {
      // Convert the log2-domain sum to natural log once per task.
      ws_log [(size_t)sample * N_DET + d] = ldsum * 0.69314718055994530942f;
      ws_sign[(size_t)sample * N_DET + d] = (sgn & 1u) ? -1.0f : 1.0f;
    }
  }
}

// Signed log-sum-exp over the 16 determinants -> complex64 (re, im) pairs.
__global__ void __launch_bounds__(256)
slater_lse_kernel(const float* __restrict__ ws_log,
                  const float* __restrict__ ws_sign,
                  float* __restrict__ out,
                  int batch)
{
  const int b = (int)(blockIdx.x * blockDim.x + threadIdx.x);
  if (b >= batch) return;

  const float* lg = ws_log  + (size_t)b * N_DET;
  const float* sg = ws_sign + (size_t)b * N_DET;

  float m = -__builtin_inff();
#pragma unroll
  for (int dd = 0; dd < N_DET; ++dd) m = fmaxf(m, lg[dd]);

  float S = 0.0f;
#pragma unroll
  for (int dd = 0; dd < N_DET; ++dd) S += sg[dd] * __expf(lg[dd] - m);

  out[2 * b + 0] = __logf(__builtin_fabsf(S)) + m;
  out[2 * b + 1] = (S < 0.0f) ? 3.14159265358979323846f : 0.0f;
}

extern "C" void kernel_launch(void* const* d_in, const int* in_sizes, int n_in,
                              void* d_out, int out_size, void* d_ws, size_t ws_size,
                              hipStream_t stream) {
  const float* U_up = (const float*)d_in[0];   // (16,128,32) f32
  const float* U_dn = (const float*)d_in[1];   // (16,128,32) f32
  const int*   y    = (const int*)d_in[2];     // (BATCH,64) i32

  const int batch = in_sizes[2] / (2 * N_EL);

  float* ws_log  = (float*)d_ws;                         // batch * N_DET
  float* ws_sign = ws_log + (size_t)batch * N_DET;       // batch * N_DET

  const int chunks = (batch + SAMPLES_PER_BLOCK - 1) / SAMPLES_PER_BLOCK;
  slater_logdet_kernel<<<dim3(chunks * N_DET), dim3(256), 0, stream>>>(
      U_up, U_dn, y, ws_log, ws_sign, batch);

  slater_lse_kernel<<<dim3((batch + 255) / 256), dim3(256), 0, stream>>>(
      ws_log, ws_sign, (float*)d_out, batch);
}